// MultiHeadedSelfAttentionWithRelPos_61658550501475
// MI455X (gfx1250) — compile-verified
//
#include <hip/hip_runtime.h>
#include <math.h>

// ---------------------------------------------------------------------------
// MultiHeadedSelfAttentionWithRelPos forward for MI455X (gfx1250, wave32).
// All GEMMs on V_WMMA_F32_16X16X4_F32 (f32 precision = reference precision).
// Rel-shift fused into the score kernel via __shfl on WMMA accumulators.
// ---------------------------------------------------------------------------

typedef __attribute__((ext_vector_type(2))) float v2f;
typedef __attribute__((ext_vector_type(8))) float v8f;

#define B_    4
#define T_    1024
#define V_    512
#define NH_   8
#define DK_   64
#define S_    2047       // 2*T-1
#define LD3V  1536       // qkv row stride (3*V)
#define MQKV  4096       // B*T
#define MPOS  8188       // B*S

__device__ __forceinline__ v8f wmma4(v2f a, v2f b, v8f c) {
  // D = A(16x4,f32) * B(4x16,f32) + C(16x16,f32)
  return __builtin_amdgcn_wmma_f32_16x16x4_f32(false, a, false, b, (short)0, c,
                                               false, false);
}

// ===========================================================================
// Kernel 1: qkv = x @ Wqkv + bqkv.   M=4096, K=512, N=1536.
// One wave -> 16x64 strip (4 accumulators, A fragment reused 4x).
// ===========================================================================
__global__ void k_qkv_gemm(const float* __restrict__ x,
                           const float* __restrict__ W,
                           const float* __restrict__ bias,
                           float* __restrict__ out) {
  const int NSTRIP = LD3V / 64;              // 24
  int id = blockIdx.x;
  int mt = id / NSTRIP, ns = id % NSTRIP;
  int m0 = mt * 16, n0 = ns * 64;
  int l = threadIdx.x & 31;
  int lane16 = l & 15, half = l >> 4;

  const float* arow = x + (size_t)(m0 + lane16) * V_;
  v8f acc[4] = {};
  for (int k0 = 0; k0 < V_; k0 += 4) {
    int ka = k0 + half * 2;
    v2f a; a.x = arow[ka]; a.y = arow[ka + 1];
#pragma unroll
    for (int j = 0; j < 4; ++j) {
      int col = n0 + j * 16 + lane16;
      v2f b;
      b.x = W[(size_t)ka * LD3V + col];
      b.y = W[(size_t)(ka + 1) * LD3V + col];
      acc[j] = wmma4(a, b, acc[j]);
    }
  }
#pragma unroll
  for (int j = 0; j < 4; ++j) {
    int col = n0 + j * 16 + lane16;
    float bv = bias[col];
#pragma unroll
    for (int r = 0; r < 8; ++r) {
      int row = m0 + r + half * 8;
      out[(size_t)row * LD3V + col] = acc[j][r] + bv;
    }
  }
}

// ===========================================================================
// Kernel 2: p = pos @ Wpos.   M=8188 (pad 8192), K=512, N=512.
// ===========================================================================
__global__ void k_pos_gemm(const float* __restrict__ pos,
                           const float* __restrict__ W,
                           float* __restrict__ out) {
  const int NSTRIP = V_ / 64;                // 8
  int id = blockIdx.x;
  int mt = id / NSTRIP, ns = id % NSTRIP;
  int m0 = mt * 16, n0 = ns * 64;
  int l = threadIdx.x & 31;
  int lane16 = l & 15, half = l >> 4;

  int am = m0 + lane16;
  bool aok = am < MPOS;
  const float* arow = pos + (size_t)am * V_;
  v8f acc[4] = {};
  for (int k0 = 0; k0 < V_; k0 += 4) {
    int ka = k0 + half * 2;
    v2f a;
    a.x = aok ? arow[ka] : 0.0f;
    a.y = aok ? arow[ka + 1] : 0.0f;
#pragma unroll
    for (int j = 0; j < 4; ++j) {
      int col = n0 + j * 16 + lane16;
      v2f b;
      b.x = W[(size_t)ka * V_ + col];
      b.y = W[(size_t)(ka + 1) * V_ + col];
      acc[j] = wmma4(a, b, acc[j]);
    }
  }
#pragma unroll
  for (int j = 0; j < 4; ++j) {
    int col = n0 + j * 16 + lane16;
#pragma unroll
    for (int r = 0; r < 8; ++r) {
      int row = m0 + r + half * 8;
      if (row < MPOS) out[(size_t)row * V_ + col] = acc[j][r];
    }
  }
}

// ===========================================================================
// Kernel 3: raw scores = (AC + rel-shifted BD) / sqrt(DK), with mask.
// One wave -> one 16x16 (t,s) tile for one (b, head).
//   AC  : (q+posu)[16xDK] @ k^T[DKx16]                      -> 16 WMMAs
//   BD  : (q+posv)[16xDK] @ p^T band of 32 rows [DKx32]     -> 32 WMMAs
//   then matrix_bd[dt][ds] = G[dt][ds + 15 - dt]  (cross-lane shuffle)
// ===========================================================================
__global__ void k_scores(const float* __restrict__ qkv,
                         const float* __restrict__ pmat,
                         const float* __restrict__ posu,
                         const float* __restrict__ posv,
                         const unsigned char* __restrict__ mask,
                         float* __restrict__ scores) {
  int id = blockIdx.x;
  int ts = id & 63; id >>= 6;
  int tt = id & 63; id >>= 6;
  int n  = id & 7;  id >>= 3;
  int b  = id;
  int t0 = tt * 16, s0 = ts * 16;

  int l = threadIdx.x & 31;
  int lane16 = l & 15, half = l >> 4;

  const float* qbase = qkv + (size_t)(b * T_) * LD3V + n * DK_;          // q
  const float* kbase = qbase + V_;                                       // k
  const float* pu = posu + n * DK_;
  const float* pv = posv + n * DK_;
  // band start: j = s + (T-1) - t ; jmin over tile (>=0 always)
  int jmin = s0 + (T_ - 1) - t0 - 15;
  const float* pband = pmat + (size_t)(b * S_) * V_ + n * DK_;

  const float* qrow = qbase + (size_t)(t0 + lane16) * LD3V;
  v8f ac = {}, g0 = {}, g1 = {};
  for (int k0 = 0; k0 < DK_; k0 += 4) {
    int ka = k0 + half * 2;
    float q0 = qrow[ka], q1 = qrow[ka + 1];
    // ---- AC ----
    v2f au; au.x = q0 + pu[ka]; au.y = q1 + pu[ka + 1];
    v2f bk;
    bk.x = kbase[(size_t)(s0 + lane16) * LD3V + ka];
    bk.y = kbase[(size_t)(s0 + lane16) * LD3V + ka + 1];
    ac = wmma4(au, bk, ac);
    // ---- BD band (two 16x16 tiles over jrel 0..31) ----
    v2f av; av.x = q0 + pv[ka]; av.y = q1 + pv[ka + 1];
    int j0 = jmin + lane16;
    int j1 = j0 + 16;
    v2f bp0, bp1;
    bp0.x = (j0 < S_) ? pband[(size_t)j0 * V_ + ka]     : 0.0f;
    bp0.y = (j0 < S_) ? pband[(size_t)j0 * V_ + ka + 1] : 0.0f;
    bp1.x = (j1 < S_) ? pband[(size_t)j1 * V_ + ka]     : 0.0f;
    bp1.y = (j1 < S_) ? pband[(size_t)j1 * V_ + ka + 1] : 0.0f;
    g0 = wmma4(av, bp0, g0);
    g1 = wmma4(av, bp1, g1);
  }

  const float scale = 0.125f;   // 1/sqrt(64)
  int ds = lane16;
  unsigned char mv = mask[b * T_ + s0 + ds];
  float* srow = scores + (((size_t)(b * NH_ + n) * T_ + t0) * T_) + s0;
#pragma unroll
  for (int r = 0; r < 8; ++r) {
    int dt = r + half * 8;
    int jrel = ds + 15 - dt;                 // 0..30
    int src = (half << 4) | (jrel & 15);     // lane owning G[dt][jrel]
    float v0 = __shfl(g0[r], src, 32);
    float v1 = __shfl(g1[r], src, 32);
    float bd = (jrel < 16) ? v0 : v1;
    float sc = (ac[r] + bd) * scale;
    if (!mv) sc = -INFINITY;
    srow[(size_t)dt * T_ + ds] = sc;
  }
}

// ===========================================================================
// Kernel 4: softmax over last dim (row length T=1024), in place.
// ===========================================================================
__global__ void k_softmax(float* __restrict__ w) {
  __shared__ float red[256];
  float* row = w + (size_t)blockIdx.x * T_;
  int t = threadIdx.x;

  float m = -INFINITY;
  for (int i = t; i < T_; i += 256) m = fmaxf(m, row[i]);
  red[t] = m; __syncthreads();
  for (int off = 128; off > 0; off >>= 1) {
    if (t < off) red[t] = fmaxf(red[t], red[t + off]);
    __syncthreads();
  }
  float rowmax = red[0]; __syncthreads();

  float s = 0.0f;
  for (int i = t; i < T_; i += 256) {
    float e = __expf(row[i] - rowmax);
    row[i] = e;
    s += e;
  }
  red[t] = s; __syncthreads();
  for (int off = 128; off > 0; off >>= 1) {
    if (t < off) red[t] += red[t + off];
    __syncthreads();
  }
  float inv = 1.0f / red[0];
  for (int i = t; i < T_; i += 256) row[i] *= inv;
}

// ===========================================================================
// Kernel 5: context = weights @ val.  Per (b, head, t-tile): 16 x 64, K=1024.
// Output layout [B, T, N*DK].
// ===========================================================================
__global__ void k_context(const float* __restrict__ w,
                          const float* __restrict__ qkv,
                          float* __restrict__ ctx) {
  int id = blockIdx.x;
  int tt = id & 63; id >>= 6;
  int n  = id & 7;  id >>= 3;
  int b  = id;
  int t0 = tt * 16;

  int l = threadIdx.x & 31;
  int lane16 = l & 15, half = l >> 4;

  const float* wrow = w + ((size_t)(b * NH_ + n) * T_ + (t0 + lane16)) * T_;
  const float* vbase = qkv + (size_t)(b * T_) * LD3V + 2 * V_ + n * DK_;

  v8f acc[4] = {};
  for (int k0 = 0; k0 < T_; k0 += 4) {
    int ka = k0 + half * 2;
    v2f a; a.x = wrow[ka]; a.y = wrow[ka + 1];
#pragma unroll
    for (int j = 0; j < 4; ++j) {
      int h = j * 16 + lane16;
      v2f bv;
      bv.x = vbase[(size_t)ka * LD3V + h];
      bv.y = vbase[(size_t)(ka + 1) * LD3V + h];
      acc[j] = wmma4(a, bv, acc[j]);
    }
  }
#pragma unroll
  for (int j = 0; j < 4; ++j) {
    int h = j * 16 + lane16;
#pragma unroll
    for (int r = 0; r < 8; ++r) {
      int t = t0 + r + half * 8;
      ctx[((size_t)(b * T_ + t)) * (NH_ * DK_) + n * DK_ + h] = acc[j][r];
    }
  }
}

// ===========================================================================
// Launch
// ===========================================================================
extern "C" void kernel_launch(void* const* d_in, const int* in_sizes, int n_in,
                              void* d_out, int out_size, void* d_ws,
                              size_t ws_size, hipStream_t stream) {
  const float*         x    = (const float*)d_in[0];
  const unsigned char* mask = (const unsigned char*)d_in[1];  // bool array
  const float*         pos  = (const float*)d_in[2];
  const float*         Wqkv = (const float*)d_in[3];
  const float*         bqkv = (const float*)d_in[4];
  const float*         Wpos = (const float*)d_in[5];
  const float*         posu = (const float*)d_in[6];
  const float*         posv = (const float*)d_in[7];

  float* ctx_out = (float*)d_out;                               // [B,T,N*DK]
  float* weights = ctx_out + (size_t)B_ * T_ * NH_ * DK_;       // [B,N,T,T]

  float* qkv  = (float*)d_ws;                                   // [4096,1536]
  float* pmat = qkv + (size_t)MQKV * LD3V;                      // [8188,512]
  (void)in_sizes; (void)n_in; (void)out_size; (void)ws_size;

  // 1) qkv projection
  k_qkv_gemm<<<(MQKV / 16) * (LD3V / 64), 32, 0, stream>>>(x, Wqkv, bqkv, qkv);
  // 2) positional projection
  k_pos_gemm<<<((MPOS + 15) / 16) * (V_ / 64), 32, 0, stream>>>(pos, Wpos, pmat);
  // 3) fused AC + rel-shifted BD scores
  k_scores<<<B_ * NH_ * 64 * 64, 32, 0, stream>>>(qkv, pmat, posu, posv, mask,
                                                  weights);
  // 4) softmax rows
  k_softmax<<<B_ * NH_ * T_, 256, 0, stream>>>(weights);
  // 5) context
  k_context<<<B_ * NH_ * 64, 32, 0, stream>>>(weights, qkv, ctx_out);
}